// BatchSemiHardTripletLoss_11407433138672
// MI455X (gfx1250) — compile-verified
//
#include <hip/hip_runtime.h>
#include <hip/hip_bf16.h>
#include <math.h>

typedef __attribute__((ext_vector_type(2))) float v2f;
typedef __attribute__((ext_vector_type(8))) float v8f;

#define MARGIN 0.2f
#define EPS_D  1e-12f
#define FINF   1e30f

// ---------------------------------------------------------------------------
// Kernel 1: row squared norms  n2[i] = sum_k E[i][k]^2
// ---------------------------------------------------------------------------
__global__ void rownorm_kernel(const float* __restrict__ E, float* __restrict__ n2,
                               int B, int D) {
    int i = blockIdx.x * blockDim.x + threadIdx.x;
    if (i >= B) return;
    const float4* row = (const float4*)(E + (size_t)i * D);
    float s = 0.f;
    int n4 = D >> 2;
    for (int k = 0; k < n4; ++k) {
        float4 v = row[k];
        s += v.x * v.x + v.y * v.y + v.z * v.z + v.w * v.w;
    }
    n2[i] = s;
}

// ---------------------------------------------------------------------------
// Kernel 2: fused Gram + distance matrix via fp32 WMMA.
// One wave32 per 16x16 output tile. D = E * E^T accumulated with
// V_WMMA_F32_16X16X4_F32, then dmat = sqrt(max(n2_i + n2_j - 2G, 0) + eps).
// ---------------------------------------------------------------------------
__global__ void gram_dmat_kernel(const float* __restrict__ E,
                                 const float* __restrict__ n2,
                                 float* __restrict__ dmat,
                                 int B, int D) {
    const int tileM = blockIdx.x * 16;
    const int tileN = blockIdx.y * 16;
    const int lane  = threadIdx.x;        // 0..31, full wave, EXEC all ones
    const int r     = lane & 15;          // row within fragment
    const int kb    = (lane >> 4) << 1;   // 0 for lanes 0-15, 2 for lanes 16-31

    const float* arow = E + (size_t)(tileM + r) * D;   // A[m][k] = E[tileM+m][k]
    const float* brow = E + (size_t)(tileN + r) * D;   // B[k][n] = E[tileN+n][k]

    v8f c = {};
#pragma unroll 4
    for (int k0 = 0; k0 < D; k0 += 4) {
        v2f a = *(const v2f*)(arow + k0 + kb);
        v2f b = *(const v2f*)(brow + k0 + kb);
        // D = A(16x4,f32) * B(4x16,f32) + C ; 8-arg form:
        // (neg_a, A, neg_b, B, c_mod, C, reuse_a, reuse_b)
        c = __builtin_amdgcn_wmma_f32_16x16x4_f32(
                false, a, false, b, (short)0, c, false, false);
    }

    // C/D layout: VGPR v, lanes 0-15 -> M=v, lanes 16-31 -> M=v+8; N = lane&15
    const int n    = lane & 15;
    const int mofs = (lane >> 4) << 3;
    const float n2n = n2[tileN + n];
#pragma unroll
    for (int v = 0; v < 8; ++v) {
        int m = v + mofs;
        float g  = c[v];
        float d2 = n2[tileM + m] + n2n - 2.0f * g;
        d2 = fmaxf(d2, 0.0f);
        dmat[(size_t)(tileM + m) * B + (tileN + n)] = sqrtf(d2 + EPS_D);
    }
}

// ---------------------------------------------------------------------------
// Kernel 3: one block (256 threads) per anchor i.
// Stage dmat row + labels into LDS; pass 1 computes min_all_neg / has_pos /
// has_neg; pass 2 does the semi-hard scan for every positive j and
// accumulates partial sums -> part[i*6 + {loss, dap, dan, active, pairs, ok}]
// ---------------------------------------------------------------------------
__global__ void anchor_pair_kernel(const float* __restrict__ dmat,
                                   const int* __restrict__ labels,
                                   float* __restrict__ part,
                                   int B) {
    extern __shared__ char smem[];
    float* rowd  = (float*)smem;                     // B floats
    int*   labs  = (int*)(rowd + B);                 // B ints
    float* sredf = (float*)(labs + B);               // 8 waves * 6 floats
    int*   sredi = (int*)(sredf + 48);               // 16 ints
    float* sbc   = (float*)(sredi + 16);             // 2 floats broadcast

    const int i    = blockIdx.x;
    const int tid  = threadIdx.x;
    const int lane = tid & 31;
    const int wid  = tid >> 5;
    const int nwav = blockDim.x >> 5;

    for (int j = tid; j < B; j += blockDim.x) {
        rowd[j] = dmat[(size_t)i * B + j];
        labs[j] = labels[j];
    }
    __syncthreads();

    const int li = labs[i];

    // ---- pass 1: min over negatives, has_pos, has_neg ----
    float minNeg = FINF;
    int hp = 0, hn = 0;
    for (int j = tid; j < B; j += blockDim.x) {
        if (labs[j] != li) { hn = 1; minNeg = fminf(minNeg, rowd[j]); }
        else if (j != i)   { hp = 1; }
    }
    for (int off = 16; off > 0; off >>= 1) {
        minNeg = fminf(minNeg, __shfl_down(minNeg, off, 32));
        hp |= __shfl_down(hp, off, 32);
        hn |= __shfl_down(hn, off, 32);
    }
    if (lane == 0) { sredf[wid] = minNeg; sredi[wid] = hp; sredi[8 + wid] = hn; }
    __syncthreads();
    if (tid == 0) {
        float m = sredf[0]; int p = sredi[0], q = sredi[8];
        for (int w = 1; w < nwav; ++w) {
            m = fminf(m, sredf[w]); p |= sredi[w]; q |= sredi[8 + w];
        }
        sbc[0] = m;
        sbc[1] = (p && q) ? 1.0f : 0.0f;
    }
    __syncthreads();
    const float minAll = sbc[0];
    const int   ok     = sbc[1] > 0.5f;

    // ---- pass 2: semi-hard negative mining per positive pair ----
    float sl = 0.f, sap = 0.f, san = 0.f, sact = 0.f, scnt = 0.f;
    if (ok) {
        for (int j = tid; j < B; j += blockDim.x) {
            if (j == i || labs[j] != li) continue;   // positives only
            const float dap = rowd[j];
            const float hi  = dap + MARGIN;
            float msemi = FINF;
            for (int k = 0; k < B; ++k) {
                float d = rowd[k];
                bool cnd = (labs[k] != li) && (d > dap) && (d < hi);
                msemi = cnd ? fminf(msemi, d) : msemi;
            }
            float dan  = (msemi < FINF) ? msemi : minAll;
            float loss = fmaxf(dap - dan + MARGIN, 0.0f);
            sl += loss; sap += dap; san += dan;
            sact += (loss > 0.0f) ? 1.0f : 0.0f;
            scnt += 1.0f;
        }
    }
    float vals[5] = {sl, sap, san, sact, scnt};
    for (int off = 16; off > 0; off >>= 1)
#pragma unroll
        for (int v = 0; v < 5; ++v)
            vals[v] += __shfl_down(vals[v], off, 32);
    __syncthreads();                // done reading sredf from pass 1
    if (lane == 0)
#pragma unroll
        for (int v = 0; v < 5; ++v) sredf[wid * 6 + v] = vals[v];
    __syncthreads();
    if (tid == 0) {
        float acc[5] = {0.f, 0.f, 0.f, 0.f, 0.f};
        for (int w = 0; w < nwav; ++w)
            for (int v = 0; v < 5; ++v) acc[v] += sredf[w * 6 + v];
        float* p = part + (size_t)i * 6;
        p[0] = acc[0]; p[1] = acc[1]; p[2] = acc[2];
        p[3] = acc[3]; p[4] = acc[4]; p[5] = ok ? 1.0f : 0.0f;
    }
}

// ---------------------------------------------------------------------------
// Kernel 4: reduce per-anchor partials -> 6 outputs
// ---------------------------------------------------------------------------
__global__ void finalize_kernel(const float* __restrict__ part,
                                float* __restrict__ out, int B) {
    __shared__ float red[8][6];
    const int tid  = threadIdx.x;
    const int lane = tid & 31;
    const int wid  = tid >> 5;

    float acc[6] = {0.f, 0.f, 0.f, 0.f, 0.f, 0.f};
    for (int a = tid; a < B; a += blockDim.x)
#pragma unroll
        for (int v = 0; v < 6; ++v) acc[v] += part[(size_t)a * 6 + v];

    for (int off = 16; off > 0; off >>= 1)
#pragma unroll
        for (int v = 0; v < 6; ++v) acc[v] += __shfl_down(acc[v], off, 32);
    if (lane == 0)
#pragma unroll
        for (int v = 0; v < 6; ++v) red[wid][v] = acc[v];
    __syncthreads();
    if (tid == 0) {
        float t[6] = {0.f, 0.f, 0.f, 0.f, 0.f, 0.f};
        int nwav = blockDim.x >> 5;
        for (int w = 0; w < nwav; ++w)
            for (int v = 0; v < 6; ++v) t[v] += red[w][v];
        float numT  = t[4];
        float denom = fmaxf(numT, 1.0f);
        out[0] = t[0] / denom;   // loss_total
        out[1] = t[1] / denom;   // mean_d_ap
        out[2] = t[2] / denom;   // mean_d_an
        out[3] = t[3] / denom;   // active_ratio
        out[4] = numT;           // num_triplets
        out[5] = t[5];           // num_anchors
    }
}

// ---------------------------------------------------------------------------
extern "C" void kernel_launch(void* const* d_in, const int* in_sizes, int n_in,
                              void* d_out, int out_size, void* d_ws, size_t ws_size,
                              hipStream_t stream) {
    const float* E      = (const float*)d_in[0];
    const int*   labels = (const int*)d_in[1];
    float*       out    = (float*)d_out;

    const int B = in_sizes[1];          // 512
    const int D = in_sizes[0] / B;      // 512

    // workspace layout (floats): n2[B] | dmat[B*B] | part[B*6]
    float* ws   = (float*)d_ws;
    float* n2   = ws;
    float* dmat = ws + B;
    float* part = ws + B + (size_t)B * B;

    rownorm_kernel<<<(B + 255) / 256, 256, 0, stream>>>(E, n2, B, D);

    dim3 tg(B / 16, B / 16);
    gram_dmat_kernel<<<tg, 32, 0, stream>>>(E, n2, dmat, B, D);

    size_t sh = (size_t)B * sizeof(float)   // rowd
              + (size_t)B * sizeof(int)     // labs
              + 48 * sizeof(float)          // sredf
              + 16 * sizeof(int)            // sredi
              + 2 * sizeof(float);          // broadcast
    anchor_pair_kernel<<<B, 256, sh, stream>>>(dmat, labels, part, B);

    finalize_kernel<<<1, 256, 0, stream>>>(part, out, B);
}